// backbone_new_NLinear_30245159698590
// MI455X (gfx1250) — compile-verified
//
#include <hip/hip_runtime.h>
#include <math.h>

typedef __attribute__((ext_vector_type(2))) float v2f;
typedef __attribute__((ext_vector_type(4))) float v4f;
typedef __attribute__((ext_vector_type(8))) float v8f;

namespace {
constexpr int B    = 32;
constexpr int C    = 21;
constexpr int T    = 512;
constexpr int L    = 336;
constexpr int FREQ = 257;                 // T/2 + 1
constexpr int BC   = B * C;               // 672 = 42*16
constexpr int HWROW = T * FREQ;           // 131584
constexpr int NT   = 8;                   // t-values per thread in k_mix
constexpr float EPS = 1e-5f;
constexpr float TWO_PI = 6.28318530717958647692f;

// workspace layout in floats
constexpr size_t WS_ZN   = 0;                          // BC*T   = 344064
constexpr size_t WS_MEAN = WS_ZN + (size_t)BC * T;     // BC
constexpr size_t WS_STD  = WS_MEAN + BC;               // BC
constexpr size_t WS_WC   = WS_STD + BC;                // FREQ*L = 86352 (stored [p][l])
constexpr size_t WS_WSN  = WS_WC + (size_t)FREQ * L;   // FREQ*L (stored [p][l])
constexpr size_t WS_MT   = WS_WSN + (size_t)FREQ * L;  // T*L (stored [t][l])
}

// ---------------------------------------------------------------------------
// K1: per-(b,c) layernorm. One block (256 thr) per row of 512.
// ---------------------------------------------------------------------------
__global__ void k_norm(const float* __restrict__ z, const float* __restrict__ aw,
                       const float* __restrict__ ab, float* __restrict__ zn,
                       float* __restrict__ meanv, float* __restrict__ stdv) {
  __shared__ float s1[256];
  __shared__ float s2[256];
  const int row = blockIdx.x;           // bc index
  const int tid = threadIdx.x;
  const float* zr = z + (size_t)row * T;
  const float x0 = zr[tid];
  const float x1 = zr[tid + 256];
  s1[tid] = x0 + x1;
  s2[tid] = x0 * x0 + x1 * x1;
  __syncthreads();
  for (int off = 128; off > 0; off >>= 1) {
    if (tid < off) { s1[tid] += s1[tid + off]; s2[tid] += s2[tid + off]; }
    __syncthreads();
  }
  const float mean = s1[0] * (1.0f / T);
  const float var  = s2[0] * (1.0f / T) - mean * mean;  // ddof=0 like jnp.var
  const float sd   = sqrtf(var + EPS);
  const float rsd  = 1.0f / sd;
  const int   c    = row % C;
  const float w    = aw[c];
  const float b    = ab[c];
  float* zo = zn + (size_t)row * T;
  zo[tid]       = (x0 - mean) * rsd * w + b;
  zo[tid + 256] = (x1 - mean) * rsd * w + b;
  if (tid == 0) { meanv[row] = mean; stdv[row] = sd; }
}

// ---------------------------------------------------------------------------
// K2: Wc[p][l] = sum_t cosb[p,t]*head_w[l, p*T+t]  (Ws with sinb = -sin).
// One wave per (l,p) pair. Each lane loads 4 float4s (global_load_b128),
// 512 B contiguous per wave per step -> streams the 177 MB head_w once at
// the HBM floor with 4x fewer vmem instructions than scalar loads.
// Trig from a 512-entry LDS table; index walks k += p (mod 512).
// ---------------------------------------------------------------------------
__global__ void k_project(const float* __restrict__ hw,
                          float* __restrict__ Wc, float* __restrict__ Wsn) {
  __shared__ float cT[512];
  __shared__ float sT[512];
  const int tid = threadIdx.x;
  for (int i = tid; i < 512; i += 256) {
    float s, c;
    sincosf(TWO_PI * (float)i * (1.0f / 512.0f), &s, &c);
    cT[i] = c; sT[i] = s;
  }
  __syncthreads();
  const int wave = tid >> 5;
  const int lane = tid & 31;
  const int pair = blockIdx.x * 8 + wave;     // = l*FREQ + p
  const int l = pair / FREQ;
  const int p = pair - l * FREQ;
  const v4f* row4 =
      (const v4f*)(hw + (size_t)l * HWROW + (size_t)p * T);
  float cr = 0.0f, ci = 0.0f;
#pragma unroll
  for (int it = 0; it < 4; ++it) {
    const v4f v = row4[it * 32 + lane];
    const int t0 = it * 128 + lane * 4;
    int k = (p * t0) & 511;
    cr += v.x * cT[k]; ci -= v.x * sT[k]; k = (k + p) & 511;
    cr += v.y * cT[k]; ci -= v.y * sT[k]; k = (k + p) & 511;
    cr += v.z * cT[k]; ci -= v.z * sT[k]; k = (k + p) & 511;
    cr += v.w * cT[k]; ci -= v.w * sT[k];
  }
  for (int off = 16; off > 0; off >>= 1) {    // wave32 reduction
    cr += __shfl_xor(cr, off);
    ci += __shfl_xor(ci, off);
  }
  if (lane == 0) {
    const float sc = (p == 0) ? 0.5f : 1.0f;  // .at[0].multiply(0.5)
    Wc [p * L + l] = cr * sc;
    Wsn[p * L + l] = ci * sc;
  }
}

// ---------------------------------------------------------------------------
// K3: Mt[t][l] = (2/T) * sum_p ( cos(2pi p t/512)*Wc[p][l] - sin*Ws[p][l] ).
// Each thread owns one l and NT=8 consecutive t values: every Wc/Ws load is
// reused 8x (cache traffic ~44 MB instead of ~354 MB). Trig lookups are
// wave-uniform (same t across lanes) -> LDS broadcast. Stores coalesced.
// Grid: T/NT = 64 blocks of 352 threads (l < 336 guarded).
// ---------------------------------------------------------------------------
__global__ void k_mix(const float* __restrict__ Wc, const float* __restrict__ Wsn,
                      float* __restrict__ Mt) {
  __shared__ float cT[512];
  __shared__ float sT[512];
  const int tid = threadIdx.x;
  for (int i = tid; i < 512; i += 352) {
    float s, c;
    sincosf(TWO_PI * (float)i * (1.0f / 512.0f), &s, &c);
    cT[i] = c; sT[i] = s;
  }
  __syncthreads();
  const int l  = tid;
  const int t0 = blockIdx.x * NT;
  if (l >= L) return;
  float acc[NT] = {};
  for (int p = 0; p < FREQ; ++p) {
    const float wc = Wc [p * L + l];
    const float ws = Wsn[p * L + l];
    int k = (p * t0) & 511;
#pragma unroll
    for (int j = 0; j < NT; ++j) {
      acc[j] += cT[k] * wc - sT[k] * ws;
      k = (k + p) & 511;
    }
  }
#pragma unroll
  for (int j = 0; j < NT; ++j)
    Mt[(size_t)(t0 + j) * L + l] = acc[j] * (2.0f / (float)T);
}

// ---------------------------------------------------------------------------
// K4: out_pre = zn (672x512) @ Mt (512x336) via V_WMMA_F32_16X16X4_F32,
// fused de-normalization epilogue. One wave per 16x16 tile; grid 42x21.
// A 16x4 layout: lane holds M=lane&15, VGPR j = K = 2*(lane>>4)+j.
// B 4x16 layout: lane holds N=lane&15, VGPR j = K = 2*(lane>>4)+j.
// D 16x16 f32:   VGPR v = row v + 8*(lane>>4), col lane&15.
// ---------------------------------------------------------------------------
__global__ void k_gemm(const float* __restrict__ zn, const float* __restrict__ Mt,
                       const float* __restrict__ meanv, const float* __restrict__ stdv,
                       const float* __restrict__ aw, const float* __restrict__ ab,
                       const float* __restrict__ hb, float* __restrict__ out) {
  const int lane = threadIdx.x;               // 0..31, full wave, EXEC all ones
  const int tm = blockIdx.x;                  // row tile (42)
  const int tn = blockIdx.y;                  // col tile (21)
  const int h  = lane >> 4;
  const int q  = lane & 15;
  const int row_m = tm * 16 + q;              // A row for this lane
  const int col_n = tn * 16 + q;              // B col for this lane
  const float* arow = zn + (size_t)row_m * T;
  v8f acc = {};
#pragma unroll 4
  for (int kk = 0; kk < T; kk += 4) {
    const int k0 = kk + 2 * h;
    v2f a, b;
    a.x = arow[k0];
    a.y = arow[k0 + 1];
    b.x = Mt[(size_t)k0 * L + col_n];
    b.y = Mt[(size_t)(k0 + 1) * L + col_n];
    acc = __builtin_amdgcn_wmma_f32_16x16x4_f32(
        /*neg_a=*/false, a, /*neg_b=*/false, b,
        /*c_mod=*/(short)0, acc, /*reuse_a=*/false, /*reuse_b=*/false);
  }
  const float hbv = hb[col_n];
  const int   r0  = tm * 16 + 8 * h;          // first of 8 consecutive rows
  int c = r0 % C;                             // single division; then wrap-add
#pragma unroll
  for (int v = 0; v < 8; ++v) {
    const int   r   = r0 + v;
    const float pre = acc[v] + hbv - ab[c];
    out[(size_t)r * L + col_n] =
        pre / (aw[c] + 1e-10f) * stdv[r] + meanv[r];
    if (++c == C) c = 0;
  }
}

// ---------------------------------------------------------------------------
extern "C" void kernel_launch(void* const* d_in, const int* in_sizes, int n_in,
                              void* d_out, int out_size, void* d_ws, size_t ws_size,
                              hipStream_t stream) {
  (void)in_sizes; (void)n_in; (void)out_size; (void)ws_size;
  const float* z  = (const float*)d_in[0];
  const float* aw = (const float*)d_in[1];
  const float* ab = (const float*)d_in[2];
  const float* hw = (const float*)d_in[3];
  const float* hb = (const float*)d_in[4];
  float* out = (float*)d_out;
  float* ws  = (float*)d_ws;

  float* zn    = ws + WS_ZN;
  float* meanv = ws + WS_MEAN;
  float* stdv  = ws + WS_STD;
  float* Wc    = ws + WS_WC;
  float* Wsn   = ws + WS_WSN;
  float* Mt    = ws + WS_MT;

  k_norm<<<BC, 256, 0, stream>>>(z, aw, ab, zn, meanv, stdv);
  k_project<<<(FREQ * L) / 8, 256, 0, stream>>>(hw, Wc, Wsn);   // 10794 blocks
  k_mix<<<T / NT, 352, 0, stream>>>(Wc, Wsn, Mt);               // 64 blocks
  k_gemm<<<dim3(BC / 16, L / 16), 32, 0, stream>>>(zn, Mt, meanv, stdv, aw, ab, hb, out);
}